// EdgeNet_11742440587663
// MI455X (gfx1250) — compile-verified
//
#include <hip/hip_runtime.h>

typedef __attribute__((ext_vector_type(2))) float v2f;
typedef __attribute__((ext_vector_type(8))) float v8f;

#define N_NODES 4096
#define E_EDGES 16384
#define F_DIM   4
#define BLOCK_THREADS 128
#define WAVES_PER_BLOCK (BLOCK_THREADS / 32)

// R(th/2) * rho * R(th/2)^T for symmetric 2x2 rho (r00, r01, r11)
__device__ __forceinline__ void rot_rho(float th, float r00, float r01, float r11,
                                        float& n00, float& n01, float& n11) {
  float h = 0.5f * th;
  float c = cosf(h), s = sinf(h);
  float cc = c * c, ss = s * s, cs = c * s;
  n00 = cc * r00 - 2.f * cs * r01 + ss * r11;
  n01 = cs * (r00 - r11) + (cc - ss) * r01;
  n11 = ss * r00 + 2.f * cs * r01 + cc * r11;
}

// Tr_control[ CNOT (sigma (x) tau) CNOT ] = p*tau + (1-p)*X tau X, p = sigma00
__device__ __forceinline__ void mix_rho(float p, float t00, float t01, float t11,
                                        float& r00, float& r01, float& r11) {
  float q = 1.f - p;
  r00 = p * t00 + q * t11;
  r01 = t01;
  r11 = p * t11 + q * t00;
}

__global__ __launch_bounds__(BLOCK_THREADS) void edgenet_ttn_kernel(
    const float* __restrict__ X, const float* __restrict__ Ri,
    const float* __restrict__ Ro, const float* __restrict__ theta,
    float* __restrict__ out) {
  // Stage X (4096 x 4 = 64KB) into LDS: reused by every K-step of every wave.
  __shared__ float Xs[N_NODES * F_DIM];
  {
    const float4* X4  = (const float4*)X;
    float4*       Xs4 = (float4*)Xs;
    for (int i = threadIdx.x; i < N_NODES; i += BLOCK_THREADS) Xs4[i] = X4[i];
  }
  __syncthreads();

  const int lane = threadIdx.x & 31;
  const int wave = threadIdx.x >> 5;
  const int half = lane >> 4;            // 0: K rows {0,1}, 1: K rows {2,3}
  const int col  = lane & 15;            // A: row m (=f); B: column n (=edge)
  const int tile = blockIdx.x * WAVES_PER_BLOCK + wave;
  const int e0   = tile * 16;

  const bool a_live = (col < F_DIM);     // A rows >= 4 are structural zeros
  const int  colc   = a_live ? col : 0;  // clamped: LDS read always in-bounds

  // B operand: B[k, n] = Ro[(n0+k)*E + e0+n]; this lane holds k = 2*half + {0,1}
  const float* roP = Ro + (size_t)(half * 2) * E_EDGES + e0 + col;
  const float* riP = Ri + (size_t)(half * 2) * E_EDGES + e0 + col;

  v8f acc_o = {};
  v8f acc_i = {};

#pragma unroll 4
  for (int n0 = 0; n0 < N_NODES; n0 += 4) {
    // Streamed exactly once: non-temporal so the 512MB stream skips L2 residency
    float bo0 = __builtin_nontemporal_load(roP);
    float bo1 = __builtin_nontemporal_load(roP + E_EDGES);
    float bi0 = __builtin_nontemporal_load(riP);
    float bi1 = __builtin_nontemporal_load(riP + E_EDGES);
    roP += (size_t)4 * E_EDGES;
    riP += (size_t)4 * E_EDGES;

    // A operand: A[m, k] = X[n0+k, m] (m = col); branch-free zero for m >= 4
    int rb = (n0 + half * 2) * F_DIM + colc;
    float x0 = Xs[rb];
    float x1 = Xs[rb + F_DIM];
    float a0 = a_live ? x0 : 0.f;
    float a1 = a_live ? x1 : 0.f;

    v2f A  = {a0, a1};
    v2f Bo = {bo0, bo1};
    v2f Bi = {bi0, bi1};

    // D = A x B + C : fp32 WMMA, exact fp32 math
    acc_o = __builtin_amdgcn_wmma_f32_16x16x4_f32(false, A, false, Bo,
                                                  (short)0, acc_o, false, false);
    acc_i = __builtin_amdgcn_wmma_f32_16x16x4_f32(false, A, false, Bi,
                                                  (short)0, acc_i, false, false);
  }

  // Lanes 0..15 now hold: acc_o[f] = bo[e0+lane, f], acc_i[f] = bi[e0+lane, f]
  if (lane < 16) {
    float phi[8];
    phi[0] = acc_o[0]; phi[1] = acc_o[1]; phi[2] = acc_o[2]; phi[3] = acc_o[3];
    phi[4] = acc_i[0]; phi[5] = acc_i[1]; phi[6] = acc_i[2]; phi[7] = acc_i[3];

    // Leaf RY(B_w) folds into first-layer RY(theta_w): angle = B_w + theta_w
    float cw[8], sw[8];
#pragma unroll
    for (int w = 0; w < 8; ++w) {
      float a = 0.5f * (phi[w] + theta[w]);
      cw[w] = cosf(a);
      sw[w] = sinf(a);
    }

    // Level 1: trace out controls 0,3,4,7 ; keep targets 1,2,5,6
    float r1_00, r1_01, r1_11, r2_00, r2_01, r2_11;
    float r5_00, r5_01, r5_11, r6_00, r6_01, r6_11;
    mix_rho(cw[0] * cw[0], cw[1] * cw[1], cw[1] * sw[1], sw[1] * sw[1], r1_00, r1_01, r1_11);
    mix_rho(cw[3] * cw[3], cw[2] * cw[2], cw[2] * sw[2], sw[2] * sw[2], r2_00, r2_01, r2_11);
    mix_rho(cw[4] * cw[4], cw[5] * cw[5], cw[5] * sw[5], sw[5] * sw[5], r5_00, r5_01, r5_11);
    mix_rho(cw[7] * cw[7], cw[6] * cw[6], cw[6] * sw[6], sw[6] * sw[6], r6_00, r6_01, r6_11);

    float s00, s01, s11, t00, t01, t11;
    // Node E: RY(th8) w1, RY(th9) w2, CNOT(1,2) -> keep wire 2
    rot_rho(theta[8], r1_00, r1_01, r1_11, s00, s01, s11);
    rot_rho(theta[9], r2_00, r2_01, r2_11, t00, t01, t11);
    mix_rho(s00, t00, t01, t11, r2_00, r2_01, r2_11);
    // Node F: RY(th10) w5, RY(th11) w6, CNOT(6,5) -> keep wire 5
    rot_rho(theta[11], r6_00, r6_01, r6_11, s00, s01, s11);
    rot_rho(theta[10], r5_00, r5_01, r5_11, t00, t01, t11);
    mix_rho(s00, t00, t01, t11, r5_00, r5_01, r5_11);
    // Node G: RY(th12) w2, RY(th13) w5, CNOT(2,5) -> keep wire 5
    rot_rho(theta[12], r2_00, r2_01, r2_11, s00, s01, s11);
    rot_rho(theta[13], r5_00, r5_01, r5_11, t00, t01, t11);
    mix_rho(s00, t00, t01, t11, r5_00, r5_01, r5_11);
    // Final RY(th14) on wire 5; output = rho[1,1] = (1 - <Z>)/2
    rot_rho(theta[14], r5_00, r5_01, r5_11, t00, t01, t11);

    out[e0 + lane] = t11;
  }
}

extern "C" void kernel_launch(void* const* d_in, const int* in_sizes, int n_in,
                              void* d_out, int out_size, void* d_ws, size_t ws_size,
                              hipStream_t stream) {
  const float* X     = (const float*)d_in[0];
  const float* Ri    = (const float*)d_in[1];
  const float* Ro    = (const float*)d_in[2];
  const float* theta = (const float*)d_in[3];
  float* out = (float*)d_out;

  // 16384 edges / 16 per wave / 4 waves per block = 256 blocks
  dim3 grid(E_EDGES / 16 / WAVES_PER_BLOCK);
  dim3 block(BLOCK_THREADS);
  edgenet_ttn_kernel<<<grid, block, 0, stream>>>(X, Ri, Ro, theta, out);
}